// SBDD_20847771254835
// MI455X (gfx1250) — compile-verified
//
#include <hip/hip_runtime.h>
#include <math.h>

#define NBR 20
#define BS 8

typedef __attribute__((ext_vector_type(2))) float v2f;
typedef __attribute__((ext_vector_type(8))) float v8f;

static __device__ __forceinline__ int imin(int a, int b) { return a < b ? a : b; }

// ===================== numpy-legacy MT19937 (RandomState(int) -> init_genrand) ==============
struct MTState { unsigned int mt[624]; int mti; };

__device__ void mt_seed(MTState* s, unsigned int seed) {
  s->mt[0] = seed;
  for (int i = 1; i < 624; ++i)
    s->mt[i] = 1812433253u * (s->mt[i - 1] ^ (s->mt[i - 1] >> 30)) + (unsigned)i;
  s->mti = 624;
}
__device__ unsigned int mt_next(MTState* s) {
  if (s->mti >= 624) {
    for (int k = 0; k < 624; ++k) {
      unsigned int y = (s->mt[k] & 0x80000000u) | (s->mt[(k + 1) % 624] & 0x7fffffffu);
      unsigned int v = s->mt[(k + 397) % 624] ^ (y >> 1);
      if (y & 1u) v ^= 0x9908b0dfu;
      s->mt[k] = v;
    }
    s->mti = 0;
  }
  unsigned int y = s->mt[s->mti++];
  y ^= (y >> 11);
  y ^= (y << 7) & 0x9d2c5680u;
  y ^= (y << 15) & 0xefc60000u;
  y ^= (y >> 18);
  return y;
}
__device__ unsigned int mt_interval(MTState* s, unsigned int mx) {  // uniform in [0,mx]
  if (mx == 0u) return 0u;
  unsigned int mask = mx;
  mask |= mask >> 1; mask |= mask >> 2; mask |= mask >> 4; mask |= mask >> 8; mask |= mask >> 16;
  unsigned int v;
  do { v = mt_next(s) & mask; } while (v > mx);
  return v;
}

__global__ void perm_kernel(int* pool1, int* pool2, int* scratch) {
  if (threadIdx.x != 0) return;
  MTState st;
  mt_seed(&st, 1u);
  for (int i = 0; i < 4096; ++i) scratch[i] = i;
  for (int i = 4095; i > 0; --i) {
    int j = (int)mt_interval(&st, (unsigned)i);
    int t = scratch[i]; scratch[i] = scratch[j]; scratch[j] = t;
  }
  for (int i = 0; i < 1024; ++i) pool1[i] = scratch[i];
  mt_seed(&st, 2u);
  for (int i = 0; i < 1024; ++i) scratch[i] = i;
  for (int i = 1023; i > 0; --i) {
    int j = (int)mt_interval(&st, (unsigned)i);
    int t = scratch[i]; scratch[i] = scratch[j]; scratch[j] = t;
  }
  for (int i = 0; i < 256; ++i) pool2[i] = scratch[i];
}

// ===================== KNN: per-point top-21 smallest squared distance ======================
__global__ void knn_kernel(const float* __restrict__ v, int N, int* __restrict__ nb) {
  __shared__ float pts[4096 * 3];  // 48 KB of the 320 KB WGP LDS
  int blocksPerB = (N + (int)blockDim.x - 1) / (int)blockDim.x;
  int b = blockIdx.x / blocksPerB;
  int nbase = (blockIdx.x % blocksPerB) * blockDim.x;
  for (int i = threadIdx.x; i < N * 3; i += blockDim.x) pts[i] = v[(long)b * N * 3 + i];
  __syncthreads();
  int n = nbase + threadIdx.x;
  if (n >= N) return;
  float px = pts[n * 3], py = pts[n * 3 + 1], pz = pts[n * 3 + 2];
  float psq = px * px + py * py + pz * pz;
  float bd[NBR + 1]; int bi[NBR + 1];
  for (int k = 0; k <= NBR; ++k) { bd[k] = INFINITY; bi[k] = -1; }
  for (int m = 0; m < N; ++m) {
    float mx = pts[m * 3], my = pts[m * 3 + 1], mz = pts[m * 3 + 2];
    float msq = mx * mx + my * my + mz * mz;
    float d = psq + msq - 2.f * (px * mx + py * my + pz * mz);
    if (d < bd[NBR]) {                    // strict <: ties keep lower index (top_k semantics)
      int p = NBR;
      while (p > 0 && bd[p - 1] > d) { bd[p] = bd[p - 1]; bi[p] = bi[p - 1]; --p; }
      bd[p] = d; bi[p] = m;
    }
  }
  for (int k = 0; k < NBR; ++k) nb[((long)b * N + n) * NBR + k] = bi[k + 1];  // drop self
}

// ===================== generic strided / batched / split-K WMMA f32 GEMM ====================
// C[z][m][n] = sum_k A[z*sAb + m*sAm + k*sAk] * B[z*sBb + k*sBk + n*sBn] (+ bias[n])
// One wave computes a 16x32 super-tile (2 accumulators, A frag reused), K unrolled x2.
// Out-of-range M rows are clamped to row M-1 (computed, never stored) -> no exec divergence.
template <bool AC>  // AC: A is K-contiguous (sAk == 1) -> aligned v2f loads
__device__ __forceinline__ void gemm_core(const float* __restrict__ Ab,
                                          const float* __restrict__ Bb,
                                          const float* __restrict__ bias,
                                          float* __restrict__ Cb,
                                          int M, int K, int N,
                                          long sAm, long sAk, long sBk, long sBn, long sCm,
                                          int ks, int ke, int m0, int n0, int lane) {
  int row = lane & 15;
  int kb  = (lane >> 4) << 1;  // ISA f32 frag: vgpr0 holds K=kb, vgpr1 holds K=kb+1
  int arow = m0 + row; if (arow > M - 1) arow = M - 1;

  const float* pa  = Ab + (long)arow * sAm + (long)(ks + kb) * sAk;
  const float* pb0 = Bb + (long)(ks + kb) * sBk + (long)(n0 + row) * sBn;
  const float* pb1 = pb0 + 16 * sBn;
  const long dA = 4 * sAk, dB = 4 * sBk;

  v8f acc0 = {0.f, 0.f, 0.f, 0.f, 0.f, 0.f, 0.f, 0.f};
  v8f acc1 = {0.f, 0.f, 0.f, 0.f, 0.f, 0.f, 0.f, 0.f};
  int kk = ks;
  for (; kk + 8 <= ke; kk += 8) {
    v2f a0, a1, b00, b01, b10, b11;
    if (AC) { a0 = *(const v2f*)pa; a1 = *(const v2f*)(pa + dA); }
    else    { a0.x = pa[0]; a0.y = pa[sAk]; a1.x = pa[dA]; a1.y = pa[dA + sAk]; }
    b00.x = pb0[0];  b00.y = pb0[sBk];
    b10.x = pb1[0];  b10.y = pb1[sBk];
    b01.x = pb0[dB]; b01.y = pb0[dB + sBk];
    b11.x = pb1[dB]; b11.y = pb1[dB + sBk];
    __builtin_prefetch(pb0 + 2 * dB, 0, 1);  // global_prefetch_b8
    pa += 2 * dA; pb0 += 2 * dB; pb1 += 2 * dB;
    acc0 = __builtin_amdgcn_wmma_f32_16x16x4_f32(false, a0, false, b00, (short)0, acc0, false, false);
    acc1 = __builtin_amdgcn_wmma_f32_16x16x4_f32(false, a0, false, b10, (short)0, acc1, false, false);
    acc0 = __builtin_amdgcn_wmma_f32_16x16x4_f32(false, a1, false, b01, (short)0, acc0, false, false);
    acc1 = __builtin_amdgcn_wmma_f32_16x16x4_f32(false, a1, false, b11, (short)0, acc1, false, false);
  }
  for (; kk < ke; kk += 4) {  // safety tail (unused: all K chunks are multiples of 16)
    v2f a0, b00, b10;
    if (AC) { a0 = *(const v2f*)pa; }
    else    { a0.x = pa[0]; a0.y = pa[sAk]; }
    b00.x = pb0[0]; b00.y = pb0[sBk];
    b10.x = pb1[0]; b10.y = pb1[sBk];
    pa += dA; pb0 += dB; pb1 += dB;
    acc0 = __builtin_amdgcn_wmma_f32_16x16x4_f32(false, a0, false, b00, (short)0, acc0, false, false);
    acc1 = __builtin_amdgcn_wmma_f32_16x16x4_f32(false, a0, false, b10, (short)0, acc1, false, false);
  }
  float bv0 = bias ? bias[n0 + row] : 0.f;
  float bv1 = bias ? bias[n0 + 16 + row] : 0.f;
  int mbase = m0 + ((lane >> 4) << 3);
  for (int r = 0; r < 8; ++r) {
    int m = mbase + r;
    if (m < M) {
      Cb[(long)m * sCm + (n0 + row)] = acc0[r] + bv0;
      Cb[(long)m * sCm + (n0 + 16 + row)] = acc1[r] + bv1;
    }
  }
}

__global__ void wmma_gemm_f32(const float* __restrict__ A, const float* __restrict__ B,
                              const float* __restrict__ bias, float* __restrict__ C,
                              int M, int K, int N,
                              long sAm, long sAk, long sAb,
                              long sBk, long sBn, long sBb,
                              long sCm, long sCb, int kLen, long sCpart) {
  int lane = threadIdx.x & 31;
  int ntn = N >> 5;  // 16x32 super-tiles along N (all N here are multiples of 32)
  int tm = blockIdx.x / ntn, tn = blockIdx.x % ntn;
  int m0 = tm << 4, n0 = tn << 5;
  const float* Ab = A + (long)blockIdx.z * sAb;
  const float* Bb = B + (long)blockIdx.z * sBb;
  float* Cb = C + (long)blockIdx.z * sCb + (long)blockIdx.y * sCpart;
  int ks = blockIdx.y * kLen;
  int ke = imin(ks + kLen, K);
  if (sAk == 1)
    gemm_core<true>(Ab, Bb, bias, Cb, M, K, N, sAm, 1, sBk, sBn, sCm, ks, ke, m0, n0, lane);
  else
    gemm_core<false>(Ab, Bb, bias, Cb, M, K, N, sAm, sAk, sBk, sBn, sCm, ks, ke, m0, n0, lane);
}

// ===================== spherical conv (surface / layer) =====================================
// mode 0: out = max_k relu(dhat.dir)               (conv_surface)
// mode 1: out = relu(fc + max_k theta*fs)          (conv_layer + relu)
// mode 2: out =      fc + max_k theta*fs           (conv_layer, last)
__global__ void conv_spatial(const float* __restrict__ v, const int* __restrict__ nb,
                             const float* __restrict__ dirs, const float* __restrict__ fo,
                             float* __restrict__ out, int N, int Fdir, int Ftot, int mode) {
  int bn = blockIdx.x;
  int b = bn / N;
  __shared__ float dh[NBR][3];
  __shared__ int nbk[NBR];
  int tid = threadIdx.x;
  if (tid < NBR) {
    int idx = nb[(long)bn * NBR + tid];
    nbk[tid] = idx;
    float dx = v[((long)b * N + idx) * 3 + 0] - v[(long)bn * 3 + 0];
    float dy = v[((long)b * N + idx) * 3 + 1] - v[(long)bn * 3 + 1];
    float dz = v[((long)b * N + idx) * 3 + 2] - v[(long)bn * 3 + 2];
    float nrm = fmaxf(sqrtf(dx * dx + dy * dy + dz * dz), 1e-12f);
    dh[tid][0] = dx / nrm; dh[tid][1] = dy / nrm; dh[tid][2] = dz / nrm;
  }
  __syncthreads();
  for (int f = tid; f < Fdir; f += blockDim.x) {
    float d0 = dirs[f], d1 = dirs[Fdir + f], d2 = dirs[2 * Fdir + f];
    float dn = fmaxf(sqrtf(d0 * d0 + d1 * d1 + d2 * d2), 1e-12f);
    d0 /= dn; d1 /= dn; d2 /= dn;
    float accm = -INFINITY;
    for (int k = 0; k < NBR; ++k) {
      float th = fmaxf(dh[k][0] * d0 + dh[k][1] * d1 + dh[k][2] * d2, 0.f);
      float val = (mode == 0) ? th : th * fo[((long)b * N + nbk[k]) * Ftot + Fdir + f];
      accm = fmaxf(accm, val);
    }
    float res;
    if (mode == 0) res = fmaxf(accm, 0.f);
    else {
      res = fo[(long)bn * Ftot + f] + accm;
      if (mode == 1) res = fmaxf(res, 0.f);
    }
    out[(long)bn * Fdir + f] = res;
  }
}

// ===================== neighbor-max pooling + point subsample ===============================
__global__ void pool_gather_max(const float* __restrict__ fm, const int* __restrict__ nb,
                                const int* __restrict__ pool, const float* __restrict__ vin,
                                float* __restrict__ out, float* __restrict__ vout,
                                int Nin, int Np, int F) {
  long t = (long)blockIdx.x * blockDim.x + threadIdx.x;
  long total = (long)BS * Np * F;
  if (t >= total) return;
  int f = (int)(t % F);
  long bi = t / F;
  int i = (int)(bi % Np);
  int b = (int)(bi / Np);
  int n = pool[i];
  float m = -INFINITY;
  for (int k = 0; k < NBR; ++k) {
    int j = nb[((long)b * Nin + n) * NBR + k];
    m = fmaxf(m, fm[((long)b * Nin + j) * F + f]);
  }
  out[((long)b * Np + i) * F + f] = m;
  if (f == 0)
    for (int c = 0; c < 3; ++c) vout[((long)b * Np + i) * 3 + c] = vin[((long)b * Nin + n) * 3 + c];
}

// ===================== BN (train mode, per-column over R rows) ==============================
__global__ void bn_columns(float* __restrict__ x, const float* __restrict__ g,
                           const float* __restrict__ bch, int R, int C) {
  int c = blockIdx.x;
  __shared__ float red[256];
  float s = 0.f;
  for (int r = threadIdx.x; r < R; r += blockDim.x) s += x[(long)r * C + c];
  red[threadIdx.x] = s; __syncthreads();
  for (int w = blockDim.x >> 1; w > 0; w >>= 1) {
    if (threadIdx.x < w) red[threadIdx.x] += red[threadIdx.x + w];
    __syncthreads();
  }
  float mean = red[0] / (float)R; __syncthreads();
  float vs = 0.f;
  for (int r = threadIdx.x; r < R; r += blockDim.x) {
    float d = x[(long)r * C + c] - mean; vs += d * d;
  }
  red[threadIdx.x] = vs; __syncthreads();
  for (int w = blockDim.x >> 1; w > 0; w >>= 1) {
    if (threadIdx.x < w) red[threadIdx.x] += red[threadIdx.x + w];
    __syncthreads();
  }
  float inv = 1.f / sqrtf(red[0] / (float)R + 1e-5f);
  float gg = g[c], bb = bch[c];
  for (int r = threadIdx.x; r < R; r += blockDim.x)
    x[(long)r * C + c] = (x[(long)r * C + c] - mean) * inv * gg + bb;
}

__global__ void softmax_rows64(float* __restrict__ x, int R) {
  __shared__ float red[64];
  int r = blockIdx.x, t = threadIdx.x;
  float v = x[(long)r * 64 + t];
  red[t] = v; __syncthreads();
  for (int w = 32; w > 0; w >>= 1) { if (t < w) red[t] = fmaxf(red[t], red[t + w]); __syncthreads(); }
  float mx = red[0]; __syncthreads();
  float e = expf(v - mx);
  red[t] = e; __syncthreads();
  for (int w = 32; w > 0; w >>= 1) { if (t < w) red[t] += red[t + w]; __syncthreads(); }
  x[(long)r * 64 + t] = e / red[0];
}

__global__ void act_sum(const float* __restrict__ act, float* __restrict__ asum) {
  int t = blockIdx.x * blockDim.x + threadIdx.x;
  if (t >= BS * 64) return;
  int b = t >> 6, k = t & 63;
  float s = 0.f;
  for (int m = 0; m < 256; ++m) s += act[((long)b * 256 + m) * 64 + k];
  asum[t] = s;
}

__global__ void vlad_sub(float* __restrict__ vlad, const float* __restrict__ asum,
                         const float* __restrict__ cw2) {
  long t = (long)blockIdx.x * blockDim.x + threadIdx.x;
  if (t >= (long)BS * 1024 * 64) return;
  int k = (int)(t & 63);
  int f = (int)((t >> 6) & 1023);
  int b = (int)(t >> 16);
  vlad[t] -= asum[b * 64 + k] * cw2[f * 64 + k];
}

__global__ void intra_norm(float* __restrict__ vlad) {  // per (b,k) L2 over f
  int bk = blockIdx.x;
  int b = bk >> 6, k = bk & 63;
  float* base = vlad + (long)b * 65536 + k;
  __shared__ float red[256];
  float s = 0.f;
  for (int f = threadIdx.x; f < 1024; f += 256) { float u = base[(long)f * 64]; s += u * u; }
  red[threadIdx.x] = s; __syncthreads();
  for (int w = 128; w > 0; w >>= 1) { if (threadIdx.x < w) red[threadIdx.x] += red[threadIdx.x + w]; __syncthreads(); }
  float inv = 1.f / fmaxf(sqrtf(red[0]), 1e-12f);
  for (int f = threadIdx.x; f < 1024; f += 256) base[(long)f * 64] *= inv;
}

__global__ void global_norm(float* __restrict__ vlad) {  // per b L2 over 65536
  int b = blockIdx.x;
  float* base = vlad + (long)b * 65536;
  __shared__ float red[1024];
  float s = 0.f;
  for (int i = threadIdx.x; i < 65536; i += 1024) { float u = base[i]; s += u * u; }
  red[threadIdx.x] = s; __syncthreads();
  for (int w = 512; w > 0; w >>= 1) { if (threadIdx.x < w) red[threadIdx.x] += red[threadIdx.x + w]; __syncthreads(); }
  float inv = 1.f / fmaxf(sqrtf(red[0]), 1e-12f);
  for (int i = threadIdx.x; i < 65536; i += 1024) base[i] *= inv;
}

__global__ void reduce_splits(const float* __restrict__ part, float* __restrict__ out, int n, int S) {
  int t = blockIdx.x * blockDim.x + threadIdx.x;
  if (t >= n) return;
  float s = 0.f;
  for (int i = 0; i < S; ++i) s += part[(long)i * n + t];  // fixed order -> deterministic
  out[t] = s;
}

__global__ void bn_rows8(const float* __restrict__ x, float* __restrict__ out,
                         const float* __restrict__ g, const float* __restrict__ bb, int C) {
  int c = blockIdx.x * blockDim.x + threadIdx.x;
  if (c >= C) return;
  float m = 0.f;
  for (int r = 0; r < BS; ++r) m += x[(long)r * C + c];
  m /= (float)BS;
  float v = 0.f;
  for (int r = 0; r < BS; ++r) { float d = x[(long)r * C + c] - m; v += d * d; }
  v /= (float)BS;
  float inv = 1.f / sqrtf(v + 1e-5f);
  for (int r = 0; r < BS; ++r) out[(long)r * C + c] = (x[(long)r * C + c] - m) * inv * g[c] + bb[c];
}

__global__ void gating_final(const float* __restrict__ vbn, const float* __restrict__ graw,
                             const float* __restrict__ gg, const float* __restrict__ gb,
                             float* __restrict__ desc) {
  int c = blockIdx.x * blockDim.x + threadIdx.x;
  if (c >= 1024) return;
  float m = 0.f;
  for (int r = 0; r < BS; ++r) m += graw[(long)r * 1024 + c];
  m /= (float)BS;
  float v = 0.f;
  for (int r = 0; r < BS; ++r) { float d = graw[(long)r * 1024 + c] - m; v += d * d; }
  v /= (float)BS;
  float inv = 1.f / sqrtf(v + 1e-5f);
  for (int r = 0; r < BS; ++r) {
    float z = (graw[(long)r * 1024 + c] - m) * inv * gg[c] + gb[c];
    float s = 1.f / (1.f + expf(-z));
    desc[(long)r * 1024 + c] = vbn[(long)r * 1024 + c] * s;
  }
}

__global__ void gem_kernel(const float* __restrict__ fm4, const float* __restrict__ p_in,
                           float* __restrict__ y) {
  int t = blockIdx.x * blockDim.x + threadIdx.x;
  if (t >= BS * 1024) return;
  int b = t >> 10, f = t & 1023;
  float p = p_in[0];
  float acc = 0.f;
  for (int m = 0; m < 256; ++m) {
    float u = fmaxf(fm4[((long)b * 256 + m) * 1024 + f], 1e-6f);
    acc += powf(u, p);
  }
  acc /= 256.f;
  y[t] = powf(acc, 1.f / p);
}

// ============================================================================================
static void launch_gemm(hipStream_t st, const float* A, const float* B, const float* bias,
                        float* C, int M, int K, int N,
                        long sAm, long sAk, long sAb, long sBk, long sBn, long sBb,
                        long sCm, long sCb, int nbatch, int nsplit, int kLen, long sCpart) {
  dim3 grid(((M + 15) / 16) * (N / 32), nsplit, nbatch);
  wmma_gemm_f32<<<grid, 32, 0, st>>>(A, B, bias, C, M, K, N, sAm, sAk, sAb,
                                     sBk, sBn, sBb, sCm, sCb, kLen, sCpart);
}

extern "C" void kernel_launch(void* const* d_in, const int* in_sizes, int n_in,
                              void* d_out, int out_size, void* d_ws, size_t ws_size,
                              hipStream_t stream) {
  (void)in_sizes; (void)n_in; (void)out_size; (void)ws_size;
  const float* x         = (const float*)d_in[0];   // (8,4096,3)
  const float* dir0      = (const float*)d_in[1];
  const float* W1        = (const float*)d_in[2];
  const float* b1        = (const float*)d_in[3];
  const float* dir1      = (const float*)d_in[4];
  const float* W2        = (const float*)d_in[5];
  const float* b2        = (const float*)d_in[6];
  const float* dir2      = (const float*)d_in[7];
  const float* W3        = (const float*)d_in[8];
  const float* b3        = (const float*)d_in[9];
  const float* dir3      = (const float*)d_in[10];
  const float* W4        = (const float*)d_in[11];
  const float* b4        = (const float*)d_in[12];
  const float* dir4      = (const float*)d_in[13];
  const float* gem_p     = (const float*)d_in[14];
  const float* cluster_w = (const float*)d_in[15];
  const float* cluster_w2= (const float*)d_in[16];
  const float* hidden_w  = (const float*)d_in[17];
  const float* bn1_g     = (const float*)d_in[18];
  const float* bn1_b     = (const float*)d_in[19];
  const float* bn2_g     = (const float*)d_in[20];
  const float* bn2_b     = (const float*)d_in[21];
  const float* gating_w  = (const float*)d_in[22];
  const float* gbn_g     = (const float*)d_in[23];
  const float* gbn_b     = (const float*)d_in[24];
  float* out = (float*)d_out;  // y (8*1024) then desc (8*1024)

  char* w = (char*)d_ws;
  auto alloc = [&](size_t bytes) -> void* {
    void* p = (void*)w;
    w += (bytes + 255) & ~(size_t)255;
    return p;
  };
  int*   permS = (int*)alloc(4096 * 4);
  int*   pool1 = (int*)alloc(1024 * 4);
  int*   pool2 = (int*)alloc(256 * 4);
  int*   nb0   = (int*)alloc((size_t)BS * 4096 * NBR * 4);
  int*   nb1   = (int*)alloc((size_t)BS * 1024 * NBR * 4);
  int*   nb2   = (int*)alloc((size_t)BS * 256 * NBR * 4);
  float* fm0   = (float*)alloc((size_t)BS * 4096 * 32 * 4);
  float* foBig = (float*)alloc((size_t)BS * 4096 * 128 * 4);  // reused for fo1 / fo3 / fo4
  float* fm1   = (float*)alloc((size_t)BS * 4096 * 64 * 4);
  float* v1    = (float*)alloc((size_t)BS * 1024 * 3 * 4);
  float* fmP1  = (float*)alloc((size_t)BS * 1024 * 64 * 4);
  float* fo2   = (float*)alloc((size_t)BS * 1024 * 256 * 4);
  float* fm2   = (float*)alloc((size_t)BS * 1024 * 128 * 4);
  float* fm3   = (float*)alloc((size_t)BS * 1024 * 256 * 4);
  float* v2    = (float*)alloc((size_t)BS * 256 * 3 * 4);
  float* fmP2  = (float*)alloc((size_t)BS * 256 * 256 * 4);
  float* fm4   = (float*)alloc((size_t)BS * 256 * 1024 * 4);
  float* act   = (float*)alloc((size_t)BS * 256 * 64 * 4);
  float* asum  = (float*)alloc((size_t)BS * 64 * 4);
  float* vlad  = (float*)alloc((size_t)BS * 1024 * 64 * 4);
  float* hidp  = (float*)alloc((size_t)64 * BS * 1024 * 4);
  float* hid   = (float*)alloc((size_t)BS * 1024 * 4);
  float* vbn   = (float*)alloc((size_t)BS * 1024 * 4);
  float* graw  = (float*)alloc((size_t)BS * 1024 * 4);
  float* fo1 = foBig; float* fo3 = foBig; float* fo4 = foBig;

  // POOL indices (deterministic, on-device)
  perm_kernel<<<1, 32, 0, stream>>>(pool1, pool2, permS);

  // ---- level 0 (4096 pts) ----
  knn_kernel<<<BS * (4096 / 256), 256, 0, stream>>>(x, 4096, nb0);
  conv_spatial<<<BS * 4096, 128, 0, stream>>>(x, nb0, dir0, nullptr, fm0, 4096, 32, 0, 0);
  launch_gemm(stream, fm0, W1, b1, fo1, BS * 4096, 32, 128, 32, 1, 0, 128, 1, 0, 128, 0, 1, 1, 32, 0);
  conv_spatial<<<BS * 4096, 128, 0, stream>>>(x, nb0, dir1, fo1, fm1, 4096, 64, 128, 1);
  pool_gather_max<<<(BS * 1024 * 64 + 255) / 256, 256, 0, stream>>>(fm1, nb0, pool1, x, fmP1, v1, 4096, 1024, 64);

  // ---- level 1 (1024 pts) ----
  knn_kernel<<<BS * (1024 / 256), 256, 0, stream>>>(v1, 1024, nb1);
  launch_gemm(stream, fmP1, W2, b2, fo2, BS * 1024, 64, 256, 64, 1, 0, 256, 1, 0, 256, 0, 1, 1, 64, 0);
  conv_spatial<<<BS * 1024, 128, 0, stream>>>(v1, nb1, dir2, fo2, fm2, 1024, 128, 256, 1);
  launch_gemm(stream, fm2, W3, b3, fo3, BS * 1024, 128, 512, 128, 1, 0, 512, 1, 0, 512, 0, 1, 1, 128, 0);
  conv_spatial<<<BS * 1024, 128, 0, stream>>>(v1, nb1, dir3, fo3, fm3, 1024, 256, 512, 1);
  pool_gather_max<<<(BS * 256 * 256 + 255) / 256, 256, 0, stream>>>(fm3, nb1, pool2, v1, fmP2, v2, 1024, 256, 256);

  // ---- level 2 (256 pts) ----
  knn_kernel<<<BS, 256, 0, stream>>>(v2, 256, nb2);
  launch_gemm(stream, fmP2, W4, b4, fo4, BS * 256, 256, 2048, 256, 1, 0, 2048, 1, 0, 2048, 0, 1, 1, 256, 0);
  conv_spatial<<<BS * 256, 128, 0, stream>>>(v2, nb2, dir4, fo4, fm4, 256, 1024, 2048, 2);

  // ---- GeM head -> y ----
  gem_kernel<<<(BS * 1024 + 255) / 256, 256, 0, stream>>>(fm4, gem_p, out);

  // ---- NetVLAD ----
  launch_gemm(stream, fm4, cluster_w, nullptr, act, BS * 256, 1024, 64, 1024, 1, 0, 64, 1, 0, 64, 0, 1, 1, 1024, 0);
  bn_columns<<<64, 256, 0, stream>>>(act, bn1_g, bn1_b, BS * 256, 64);
  softmax_rows64<<<BS * 256, 64, 0, stream>>>(act, BS * 256);
  act_sum<<<(BS * 64 + 255) / 256, 256, 0, stream>>>(act, asum);
  // vlad[b,f,k] = sum_m fm4[b,m,f] * act[b,m,k]   (A strided: sAk = 1024)
  launch_gemm(stream, fm4, act, nullptr, vlad, 1024, 256, 64,
              1, 1024, (long)256 * 1024, 64, 1, (long)256 * 64, 64, (long)1024 * 64, BS, 1, 256, 0);
  vlad_sub<<<(BS * 1024 * 64 + 255) / 256, 256, 0, stream>>>(vlad, asum, cluster_w2);
  intra_norm<<<BS * 64, 256, 0, stream>>>(vlad);
  global_norm<<<BS, 1024, 0, stream>>>(vlad);
  // hidden GEMM: (8 x 65536) @ (65536 x 1024), split-K x64 for occupancy, deterministic reduce
  launch_gemm(stream, vlad, hidden_w, nullptr, hidp, BS, 65536, 1024,
              65536, 1, 0, 1024, 1, 0, 1024, 0, 1, 64, 1024, (long)BS * 1024);
  reduce_splits<<<(BS * 1024 + 255) / 256, 256, 0, stream>>>(hidp, hid, BS * 1024, 64);
  bn_rows8<<<(1024 + 255) / 256, 256, 0, stream>>>(hid, vbn, bn2_g, bn2_b, 1024);
  launch_gemm(stream, vbn, gating_w, nullptr, graw, BS, 1024, 1024, 1024, 1, 0, 1024, 1, 0, 1024, 0, 1, 1, 1024, 0);
  gating_final<<<(1024 + 255) / 256, 256, 0, stream>>>(vbn, graw, gbn_g, gbn_b, out + BS * 1024);
}